// GeneticPathPlanner_13314398618135
// MI455X (gfx1250) — compile-verified
//
#include <hip/hip_runtime.h>
#include <hip/hip_bf16.h>

#define POP       4096
#define LEN       2048
#define GRID      512
#define NTHREADS  1024
#define NWAVES    (NTHREADS / 32)
#define BITWORDS  ((GRID * GRID) / 32)   /* 8192 dwords = 32 KB per bitmap */

/* gfx1250 async global->LDS path (ASYNCcnt), guarded so the kernel always compiles */
#if defined(__has_builtin)
#  if __has_builtin(__builtin_amdgcn_global_load_async_to_lds_b128) && \
      __has_builtin(__builtin_amdgcn_s_wait_asynccnt)
#    define USE_ASYNC_LDS 1
#  endif
#endif
#ifndef USE_ASYNC_LDS
#  define USE_ASYNC_LDS 0
#endif

typedef int v4i __attribute__((ext_vector_type(4)));
typedef __attribute__((address_space(1))) v4i* gv4i_ptr;   /* global v4i32* */
typedef __attribute__((address_space(3))) v4i* lv4i_ptr;   /* LDS    v4i32* */

__device__ __forceinline__ int l1_gt1(int a, int b) {
    int ax = a >> 9, ay = a & (GRID - 1);
    int bx = b >> 9, by = b & (GRID - 1);
    int dx = ax > bx ? ax - bx : bx - ax;
    int dy = ay > by ? ay - by : by - ay;
    return (dx + dy > 1) ? 1 : 0;
}

__global__ __launch_bounds__(NTHREADS, 1)
void GeneticPathPlanner_fitness_kernel(const float* __restrict__ population,
                                       const float* __restrict__ target_prob,
                                       float* __restrict__ out)
{
    /* Exactly 80 KB LDS: 2 blocks fit a 160 KB WGP budget (4 in 320 KB). */
    __shared__ __align__(16) unsigned int s_seen[BITWORDS];   /* 32 KB visited bitmap   */
    __shared__ __align__(16) unsigned int s_dup [BITWORDS];   /* 32 KB revisited bitmap */
    __shared__ __align__(16) float        s_stage[LEN * 2];   /* 16 KB coord stage;
                                                                 words [0,1024): even codes
                                                                 words [1024,1088): partials
                                                                 (both reused after reads) */

    const int t    = threadIdx.x;
    const int lane = t & 31;
    const int wv   = t >> 5;
    const int p    = blockIdx.x;
    const float* gsrc = population + (size_t)p * (LEN * 2);

    /* ---- Phase 0: async copy of this path's 16 KB of coords into LDS (ASYNCcnt).
       Thread t moves bytes [16t, 16t+16) -> covers all LEN*2 floats exactly.       */
#if USE_ASYNC_LDS
    {
        unsigned lds_off = (unsigned)(uintptr_t)(&s_stage[t * 4]);   /* DS byte offset */
        lv4i_ptr laddr = (lv4i_ptr)(uintptr_t)lds_off;
        gv4i_ptr gaddr = (gv4i_ptr)(uintptr_t)(gsrc + (size_t)t * 4);
        __builtin_amdgcn_global_load_async_to_lds_b128(gaddr, laddr, 0, 0);
    }
#else
    float4 v = ((const float4*)gsrc)[t];
#endif

    /* ---- Phase 1: zero both bitmaps (b128 stores, lane-contiguous), overlapped. ---- */
    {
        uint4* sv = (uint4*)s_seen;   /* 2048 uint4 */
        uint4* dv = (uint4*)s_dup;    /* 2048 uint4 */
        uint4  z  = make_uint4(0u, 0u, 0u, 0u);
        sv[t] = z;  sv[t + NTHREADS] = z;
        dv[t] = z;  dv[t + NTHREADS] = z;
    }

#if USE_ASYNC_LDS
    __builtin_amdgcn_s_wait_asynccnt(0);
#else
    *(float4*)&s_stage[t * 4] = v;
#endif
    __syncthreads();                                   /* barrier #1 */

    /* ---- Phase 2: codes, reward gathers, bitmap membership (ds_or_rtn). ---- */
    float4 pt = *(const float4*)&s_stage[t * 4];       /* points 2t and 2t+1 */
    int c0 = (int)pt.x * GRID + (int)pt.y;
    int c1 = (int)pt.z * GRID + (int)pt.w;

    float reward = target_prob[c0] + target_prob[c1];  /* L2-resident random gathers */

    {
        unsigned w = (unsigned)c0 >> 5, b = 1u << (c0 & 31);
        unsigned old = atomicOr(&s_seen[w], b);        /* ds_or_rtn_b32 */
        if (old & b) atomicOr(&s_dup[w], b);           /* idempotent dup mark */
    }
    {
        unsigned w = (unsigned)c1 >> 5, b = 1u << (c1 & 31);
        unsigned old = atomicOr(&s_seen[w], b);
        if (old & b) atomicOr(&s_dup[w], b);
    }
    __syncthreads();                                   /* barrier #2: stage reads + atomics done */

    /* ---- Phase 3: publish even-point codes (overlay stage words [0,1024)). ---- */
    unsigned* s_evn = (unsigned*)s_stage;
    s_evn[t] = (unsigned)c0;
    __syncthreads();                                   /* barrier #3 */

    /* ---- Phase 4: continuity (step 2t from regs, step 2t+1 via one LDS read)
            + dup-bitmap popcount. ---- */
    int cont = l1_gt1(c0, c1);                         /* step 2t, always valid */
    if (t < NTHREADS - 1)
        cont += l1_gt1(c1, (int)s_evn[t + 1]);         /* step 2t+1 */

    int dupc = 0;
    #pragma unroll
    for (int j = 0; j < BITWORDS / NTHREADS; ++j)      /* 8 words, lane-contiguous */
        dupc += __popc(s_dup[t + j * NTHREADS]);

    /* ---- Phase 5: deterministic wave32 butterfly reduction. ---- */
    unsigned packed = (unsigned)cont | ((unsigned)dupc << 16);  /* sums fit 16b each */
    #pragma unroll
    for (int m = 16; m > 0; m >>= 1) {
        reward += __shfl_xor(reward, m, 32);
        packed += __shfl_xor(packed, m, 32);
    }

    float*    rpf = &s_stage[NTHREADS];                /* 32 wave partials (floats) */
    unsigned* rpi = (unsigned*)&s_stage[NTHREADS + NWAVES];
    if (lane == 0) { rpf[wv] = reward; rpi[wv] = packed; }
    __syncthreads();                                   /* barrier #4 */

    if (t < NWAVES) {
        float    fr = rpf[t];
        unsigned pr = rpi[t];
        #pragma unroll
        for (int m = 16; m > 0; m >>= 1) {
            fr += __shfl_xor(fr, m, 32);
            pr += __shfl_xor(pr, m, 32);
        }
        if (t == 0)
            out[p] = fr - 0.5f * (float)(pr & 0xFFFFu) - 0.2f * (float)(pr >> 16);
    }
}

extern "C" void kernel_launch(void* const* d_in, const int* in_sizes, int n_in,
                              void* d_out, int out_size, void* d_ws, size_t ws_size,
                              hipStream_t stream)
{
    (void)in_sizes; (void)n_in; (void)out_size; (void)d_ws; (void)ws_size;
    const float* population  = (const float*)d_in[0];   /* [POP, LEN, 2] f32 */
    const float* target_prob = (const float*)d_in[1];   /* [GRID, GRID] f32  */
    float*       out         = (float*)d_out;           /* [POP] f32         */

    GeneticPathPlanner_fitness_kernel<<<POP, NTHREADS, 0, stream>>>(population, target_prob, out);
}